// MambaModel_4793183502530
// MI455X (gfx1250) — compile-verified
//
#include <hip/hip_runtime.h>
#include <math.h>

#define N_LAYERS 6
#define D_MODEL  272
#define D_STATE  64
#define D_INNER  544
#define D_CONV   4
#define DT_RANK  17
#define NPROJ    (DT_RANK + 2 * D_STATE)   // 145
#define BB       4
#define LL       2048
#define NTOK     (BB * LL)                 // 8192

#define GMT 2                 // 16-row tiles per wave (M)
#define GNT 4                 // 16-col tiles per wave (N)
#define SCAN_DPB 8            // d-channels per scan block (8 waves)
#define SCAN_TC  16           // timesteps staged per LDS chunk

#if defined(__gfx1250__) && __has_builtin(__builtin_amdgcn_global_load_async_to_lds_b32) && \
    __has_builtin(__builtin_amdgcn_s_wait_asynccnt)
#define HAVE_ASYNC_LDS 1
#else
#define HAVE_ASYNC_LDS 0
#endif

typedef __attribute__((ext_vector_type(2))) float v2f;
typedef __attribute__((ext_vector_type(8))) float v8f;

__device__ __forceinline__ float warp_sum(float v) {
  #pragma unroll
  for (int off = 16; off > 0; off >>= 1) v += __shfl_xor(v, off, 32);
  return v;
}

__device__ __forceinline__ float silu_f(float v) { return v / (1.0f + __expf(-v)); }

// ---------------------------------------------------------------- LayerNorm
__global__ void ln_kernel(const float* __restrict__ x, const float* __restrict__ w,
                          const float* __restrict__ b, float* __restrict__ o) {
  int lane = threadIdx.x & 31;
  int tok  = blockIdx.x * (blockDim.x >> 5) + (threadIdx.x >> 5);
  const float* xr = x + (size_t)tok * D_MODEL;
  float s = 0.f, s2 = 0.f;
  for (int i = lane; i < D_MODEL; i += 32) { float v = xr[i]; s += v; s2 += v * v; }
  s  = warp_sum(s);
  s2 = warp_sum(s2);
  float mu  = s * (1.0f / D_MODEL);
  float var = s2 * (1.0f / D_MODEL) - mu * mu;
  float inv = rsqrtf(var + 1e-5f);
  float* orow = o + (size_t)tok * D_MODEL;
  for (int i = lane; i < D_MODEL; i += 32)
    orow[i] = (xr[i] - mu) * inv * w[i] + b[i];
}

// ---------------------------------------------------------------- WMMA GEMM
// O[m,n] = act( sum_k A[m*lda+k] * W[n*K+k] + bias[n] ) (+ resid[m*ldr+n])
// One wave computes a 32x64 macro-tile (GMT x GNT 16x16 WMMA tiles).
// Out-of-range N columns are CLAMPED for loads (columns are independent in
// D = A*B) so the k-loop is branch-free with full EXEC; stores are guarded.
// act: 0 = none, 2 = softplus
__global__ void wmma_gemm(const float* __restrict__ A, int lda,
                          const float* __restrict__ W,
                          const float* __restrict__ bias,
                          const float* __restrict__ resid, int ldr,
                          float* __restrict__ O, int ldo,
                          int M, int N, int K, int act) {
  const int lane = threadIdx.x & 31;
  const int half = lane >> 4;      // 0: K pair {0,1}; 1: K pair {2,3}
  const int l16  = lane & 15;
  const int m0 = blockIdx.y * (16 * GMT);
  const int n0 = blockIdx.x * (16 * GNT);

  const float* arow[GMT];
  #pragma unroll
  for (int mt = 0; mt < GMT; ++mt)
    arow[mt] = A + (size_t)(m0 + mt * 16 + l16) * lda;   // rows always valid

  const float* wrow[GNT];
  int ncol[GNT];
  #pragma unroll
  for (int nt = 0; nt < GNT; ++nt) {
    int n = n0 + nt * 16 + l16;
    ncol[nt] = n;
    int ncl = (n < N) ? n : (N - 1);                     // clamp for loads
    wrow[nt] = W + (size_t)ncl * K;
  }

  v8f acc[GMT][GNT];
  #pragma unroll
  for (int nt = 0; nt < GNT; ++nt) {
    int ncl = (ncol[nt] < N) ? ncol[nt] : (N - 1);
    float binit = bias ? bias[ncl] : 0.f;
    #pragma unroll
    for (int mt = 0; mt < GMT; ++mt)
      #pragma unroll
      for (int r = 0; r < 8; ++r) acc[mt][nt][r] = binit;
  }

  const int kbase = half << 1;
  int k0 = 0;
  for (; k0 + 4 <= K; k0 += 4) {
    int kA = k0 + kbase;
    v2f af[GMT], bf[GNT];
    #pragma unroll
    for (int mt = 0; mt < GMT; ++mt) {
      af[mt].x = arow[mt][kA];
      af[mt].y = arow[mt][kA + 1];
    }
    #pragma unroll
    for (int nt = 0; nt < GNT; ++nt) {
      bf[nt].x = wrow[nt][kA];
      bf[nt].y = wrow[nt][kA + 1];
    }
    __builtin_prefetch(arow[0] + kA + 32, 0, 3);
    #pragma unroll
    for (int mt = 0; mt < GMT; ++mt)
      #pragma unroll
      for (int nt = 0; nt < GNT; ++nt)
        acc[mt][nt] = __builtin_amdgcn_wmma_f32_16x16x4_f32(
            false, af[mt], false, bf[nt], (short)0, acc[mt][nt], false, false);
  }
  if (k0 < K) {  // K remainder (dt_proj K=17): zero-padded fringe
    int kA = k0 + kbase;
    v2f af[GMT], bf[GNT];
    #pragma unroll
    for (int mt = 0; mt < GMT; ++mt) {
      af[mt].x = (kA     < K) ? arow[mt][kA]     : 0.f;
      af[mt].y = (kA + 1 < K) ? arow[mt][kA + 1] : 0.f;
    }
    #pragma unroll
    for (int nt = 0; nt < GNT; ++nt) {
      bf[nt].x = (kA     < K) ? wrow[nt][kA]     : 0.f;
      bf[nt].y = (kA + 1 < K) ? wrow[nt][kA + 1] : 0.f;
    }
    #pragma unroll
    for (int mt = 0; mt < GMT; ++mt)
      #pragma unroll
      for (int nt = 0; nt < GNT; ++nt)
        acc[mt][nt] = __builtin_amdgcn_wmma_f32_16x16x4_f32(
            false, af[mt], false, bf[nt], (short)0, acc[mt][nt], false, false);
  }

  #pragma unroll
  for (int nt = 0; nt < GNT; ++nt) {
    int n = ncol[nt];
    if (n < N) {
      #pragma unroll
      for (int mt = 0; mt < GMT; ++mt) {
        #pragma unroll
        for (int r = 0; r < 8; ++r) {
          int m = m0 + mt * 16 + r + (half << 3);
          float v = acc[mt][nt][r];
          if (act == 2) v = (v > 20.f) ? v : log1pf(__expf(v));   // softplus
          if (resid) v += resid[(size_t)m * ldr + n];
          O[(size_t)m * ldo + n] = v;
        }
      }
    }
  }
}

// ------------------------------------------------- causal depthwise conv + silu
__global__ void conv_silu_kernel(const float* __restrict__ xz,
                                 const float* __restrict__ cw,
                                 const float* __restrict__ cb,
                                 float* __restrict__ xc) {
  int idx = blockIdx.x * blockDim.x + threadIdx.x;
  if (idx >= NTOK * D_INNER) return;
  int d   = idx % D_INNER;
  int tok = idx / D_INNER;
  int t   = tok % LL;
  float acc = cb[d];
  const float* base = xz + (size_t)tok * (2 * D_INNER) + d;
  #pragma unroll
  for (int k = 0; k < D_CONV; ++k) {
    int tt = t - (D_CONV - 1) + k;
    if (tt >= 0)
      acc += base[(ptrdiff_t)(k - (D_CONV - 1)) * (2 * D_INNER)] * cw[d * D_CONV + k];
  }
  xc[idx] = silu_f(acc);
}

// ---------------------------------------------------------------- gating
__global__ void gate_kernel(const float* __restrict__ y, const float* __restrict__ xz,
                            float* __restrict__ yg) {
  int idx = blockIdx.x * blockDim.x + threadIdx.x;
  if (idx >= NTOK * D_INNER) return;
  int d   = idx % D_INNER;
  int tok = idx / D_INNER;
  float z = xz[(size_t)tok * (2 * D_INNER) + D_INNER + d];
  yg[idx] = y[idx] * silu_f(z);
}

// ---------------------------------------------------------------- selective scan
// Block = 8 waves = 8 d-channels of one batch. B_t/C_t (128 floats/t) are
// staged into LDS in 16-timestep chunks, double-buffered: on gfx1250 the
// staging uses GLOBAL_LOAD_ASYNC_TO_LDS_B32 (ASYNCcnt) so chunk c+1 streams
// in while the recurrence processes chunk c.
__global__ void scan_kernel(const float* __restrict__ xc,     // u     (NTOK, D_INNER)
                            const float* __restrict__ delta,  //       (NTOK, D_INNER)
                            const float* __restrict__ dbl,    //       (NTOK, NPROJ)
                            const float* __restrict__ A_log,  // (D_INNER, D_STATE)
                            const float* __restrict__ Dskip,  // (D_INNER)
                            float* __restrict__ y) {
  __shared__ float sbc[2][SCAN_TC * 2 * D_STATE];   // 2 x 8KB

  const int nblk_d = D_INNER / SCAN_DPB;            // 68
  int b  = blockIdx.x / nblk_d;
  int d0 = (blockIdx.x % nblk_d) * SCAN_DPB;
  int w    = threadIdx.x >> 5;
  int lane = threadIdx.x & 31;
  int d = d0 + w;

  float a0 = -__expf(A_log[d * D_STATE + lane]);
  float a1 = -__expf(A_log[d * D_STATE + lane + 32]);
  float Dv = Dskip[d];
  const float* dp = delta + (size_t)b * LL * D_INNER + d;
  const float* up = xc    + (size_t)b * LL * D_INNER + d;
  const float* bc = dbl   + (size_t)b * LL * NPROJ + DT_RANK;  // B@0, C@64 within row
  float* yp = y + (size_t)b * LL * D_INNER + d;
  float h0 = 0.f, h1 = 0.f;

  const int tid = threadIdx.x;
  auto stage = [&](int buf, int t0) {
    #pragma unroll
    for (int e = 0; e < (SCAN_TC * 128) / 256; ++e) {
      int idx = tid + e * 256;
      int row = idx >> 7, col = idx & 127;
      const float* g = bc + (size_t)(t0 + row) * NPROJ + col;
#if HAVE_ASYNC_LDS
      __builtin_amdgcn_global_load_async_to_lds_b32(
          (__attribute__((address_space(1))) int*)(void*)g,
          (__attribute__((address_space(3))) int*)(void*)&sbc[buf][idx],
          0, 0);
#else
      sbc[buf][idx] = *g;
#endif
    }
  };

  stage(0, 0);
#if HAVE_ASYNC_LDS
  __builtin_amdgcn_s_wait_asynccnt(0);
#endif
  __syncthreads();

  const int nchunk = LL / SCAN_TC;
  for (int c = 0; c < nchunk; ++c) {
    if (c + 1 < nchunk) stage((c + 1) & 1, (c + 1) * SCAN_TC);
    const float* sb = &sbc[c & 1][0];
    int t0 = c * SCAN_TC;
    #pragma unroll 4
    for (int j = 0; j < SCAN_TC; ++j) {
      int t = t0 + j;
      float dt = dp[(size_t)t * D_INNER];
      float ut = up[(size_t)t * D_INNER];
      const float* r = sb + j * 128;
      float Bv0 = r[lane];
      float Bv1 = r[lane + 32];
      float Cv0 = r[64 + lane];
      float Cv1 = r[96 + lane];
      float du = dt * ut;
      h0 = __expf(dt * a0) * h0 + du * Bv0;
      h1 = __expf(dt * a1) * h1 + du * Bv1;
      float p = warp_sum(h0 * Cv0 + h1 * Cv1);
      if (lane == 0) yp[(size_t)t * D_INNER] = p + ut * Dv;
    }
#if HAVE_ASYNC_LDS
    __builtin_amdgcn_s_wait_asynccnt(0);
#endif
    __syncthreads();
  }
}

// ---------------------------------------------------------------- head
__global__ void mean_kernel(const float* __restrict__ x, float* __restrict__ feat) {
  int b = blockIdx.x;
  int j = threadIdx.x;
  if (j >= D_MODEL) return;
  const float* base = x + (size_t)b * LL * D_MODEL + j;
  float s = 0.f;
  for (int t = 0; t < LL; ++t) s += base[(size_t)t * D_MODEL];
  feat[b * D_MODEL + j] = s * (1.0f / LL);
}

__global__ void score_kernel(const float* __restrict__ feat, const float* __restrict__ rw,
                             const float* __restrict__ rb, float* __restrict__ scores) {
  int tid = threadIdx.x;
  if (tid >= BB * 4) return;
  int b = tid >> 2, s = tid & 3;
  float acc = rb[s];
  for (int j = 0; j < D_MODEL; ++j) acc += feat[b * D_MODEL + j] * rw[s * D_MODEL + j];
  scores[b * 4 + s] = acc;
}

// ---------------------------------------------------------------- host
extern "C" void kernel_launch(void* const* d_in, const int* in_sizes, int n_in,
                              void* d_out, int out_size, void* d_ws, size_t ws_size,
                              hipStream_t stream) {
  (void)in_sizes; (void)n_in; (void)out_size; (void)ws_size;
  const float* x_in  = (const float*)d_in[0];
  const float* ln_w  = (const float*)d_in[1];
  const float* ln_b  = (const float*)d_in[2];
  const float* Wi    = (const float*)d_in[3];
  const float* cw    = (const float*)d_in[4];
  const float* cb    = (const float*)d_in[5];
  const float* Wx    = (const float*)d_in[6];
  const float* Wdt   = (const float*)d_in[7];
  const float* bdt   = (const float*)d_in[8];
  const float* A_log = (const float*)d_in[9];
  const float* Dskip = (const float*)d_in[10];
  const float* Wo    = (const float*)d_in[11];
  const float* reg_w = (const float*)d_in[12];
  const float* reg_b = (const float*)d_in[13];
  float* out = (float*)d_out;

  float* ws = (float*)d_ws;
  size_t off = 0;
  auto alloc = [&](size_t n) { float* p = ws + off; off += n; return p; };
  float* xA  = alloc((size_t)NTOK * D_MODEL);
  float* xB  = alloc((size_t)NTOK * D_MODEL);
  float* lnb = alloc((size_t)NTOK * D_MODEL);
  float* xz  = alloc((size_t)NTOK * 2 * D_INNER);
  float* xc  = alloc((size_t)NTOK * D_INNER);
  float* dbl = alloc((size_t)NTOK * NPROJ);
  float* dl  = alloc((size_t)NTOK * D_INNER);
  float* yb  = alloc((size_t)NTOK * D_INNER);
  float* yg  = alloc((size_t)NTOK * D_INNER);

  (void)hipMemcpyAsync(xA, x_in, sizeof(float) * (size_t)NTOK * D_MODEL,
                       hipMemcpyDeviceToDevice, stream);

  float* cur = xA;
  float* nxt = xB;
  const int TMG = NTOK / (16 * GMT);                        // 256 M tile-groups
  auto ngrp = [](int N) { return ((N + 15) / 16 + GNT - 1) / GNT; };
  const int NEL = NTOK * D_INNER;

  for (int i = 0; i < N_LAYERS; ++i) {
    // 1. layernorm
    ln_kernel<<<NTOK / 8, 256, 0, stream>>>(cur, ln_w + (size_t)i * D_MODEL,
                                            ln_b + (size_t)i * D_MODEL, lnb);
    // 2. in_proj: xz = ln @ Wi^T   (M=8192, N=1088, K=272)
    wmma_gemm<<<dim3(ngrp(2 * D_INNER), TMG), 32, 0, stream>>>(
        lnb, D_MODEL, Wi + (size_t)i * 2 * D_INNER * D_MODEL, nullptr, nullptr, 0,
        xz, 2 * D_INNER, NTOK, 2 * D_INNER, D_MODEL, 0);
    // 3. causal depthwise conv + silu -> xc
    conv_silu_kernel<<<(NEL + 255) / 256, 256, 0, stream>>>(
        xz, cw + (size_t)i * D_INNER * D_CONV, cb + (size_t)i * D_INNER, xc);
    // 4. x_proj: dbl = xc @ Wx^T   (N=145)
    wmma_gemm<<<dim3(ngrp(NPROJ), TMG), 32, 0, stream>>>(
        xc, D_INNER, Wx + (size_t)i * NPROJ * D_INNER, nullptr, nullptr, 0,
        dbl, NPROJ, NTOK, NPROJ, D_INNER, 0);
    // 5. delta = softplus(dbl[:, :17] @ Wdt^T + bdt)   (K=17)
    wmma_gemm<<<dim3(ngrp(D_INNER), TMG), 32, 0, stream>>>(
        dbl, NPROJ, Wdt + (size_t)i * D_INNER * DT_RANK, bdt + (size_t)i * D_INNER,
        nullptr, 0, dl, D_INNER, NTOK, D_INNER, DT_RANK, 2);
    // 6. selective scan (LDS-staged, async double-buffered)
    scan_kernel<<<BB * (D_INNER / SCAN_DPB), 32 * SCAN_DPB, 0, stream>>>(
        xc, dl, dbl, A_log + (size_t)i * D_INNER * D_STATE,
        Dskip + (size_t)i * D_INNER, yb);
    // 7. yg = y * silu(z)
    gate_kernel<<<(NEL + 255) / 256, 256, 0, stream>>>(yb, xz, yg);
    // 8. nxt = cur + yg @ Wo^T
    wmma_gemm<<<dim3(ngrp(D_MODEL), TMG), 32, 0, stream>>>(
        yg, D_INNER, Wo + (size_t)i * D_MODEL * D_INNER, nullptr, cur, D_MODEL,
        nxt, D_MODEL, NTOK, D_MODEL, D_INNER, 0);
    float* tmp = cur; cur = nxt; nxt = tmp;
  }

  // head: feat = mean over L; scores = feat @ reg_w^T + reg_b
  float* feat = out + BB * 4;  // d_out layout: scores (4x4) then feat (4x272)
  mean_kernel<<<BB, 288, 0, stream>>>(cur, feat);
  score_kernel<<<1, 32, 0, stream>>>(feat, reg_w, reg_b, out);
}